// GCN_Conv_91216515432980
// MI455X (gfx1250) — compile-verified
//
#include <hip/hip_runtime.h>
#include <cstdint>

#define NNODES 50000
#define NEDGES 800000
#define DFEAT  64

typedef float v2f __attribute__((ext_vector_type(2)));
typedef float v8f __attribute__((ext_vector_type(8)));

__device__ __forceinline__ void atomAddF(float* p, float v) {
  // lowers to global_atomic_add_f32 (no CAS loop) — L2-resident atomics
  unsafeAtomicAdd(p, v);
}

// ---------------- degree / normalization ----------------

__global__ void k_set1(float* __restrict__ deg, int n) {
  int i = blockIdx.x * blockDim.x + threadIdx.x;
  if (i < n) deg[i] = 1.0f;  // self-loop weight
}

__global__ void k_degcount(const int* __restrict__ ei, float* __restrict__ deg, int nE) {
  int e = blockIdx.x * blockDim.x + threadIdx.x;
  if (e < nE) atomAddF(&deg[ei[nE + e]], 1.0f);  // col = targets
}

__global__ void k_rsqrt_inplace(float* __restrict__ d, int n) {
  int i = blockIdx.x * blockDim.x + threadIdx.x;
  if (i < n) {
    float v = d[i];
    d[i] = (v > 0.0f) ? rsqrtf(v) : 0.0f;
  }
}

// ---------------- dense GEMM via fp32 WMMA ----------------
// H[50000x64] = X[50000x64] @ W[64x64], one 16x16 tile per wave,
// K-loop of 16 x V_WMMA_F32_16X16X4_F32. 50000 % 16 == 0 -> no guards,
// EXEC stays all-ones as WMMA requires.
__global__ void __launch_bounds__(128) k_gemm_wmma(const float* __restrict__ X,
                                                   const float* __restrict__ W,
                                                   float* __restrict__ H) {
  const int lane  = threadIdx.x;     // 0..31 (wave32)
  const int col   = lane & 15;       // N-col for B/C/D, M-row for A
  const int khalf = lane >> 4;       // lanes 16..31 hold K+2 (A/B), M+8 (C/D)
  const int mBase = blockIdx.x * 16;
  const int nBase = threadIdx.y * 16;  // 4 waves cover the 4 N-tiles of D=64

  const float* __restrict__ arow = X + (size_t)(mBase + col) * DFEAT;

  v8f acc = {};
#pragma unroll
  for (int k0 = 0; k0 < DFEAT; k0 += 4) {
    const int ka = k0 + khalf * 2;
    v2f a;
    a.x = arow[ka];                                  // A: VGPR0 = K, VGPR1 = K+1
    a.y = arow[ka + 1];
    v2f b;
    b.x = W[(size_t)ka * DFEAT + nBase + col];       // B: row K striped over lanes
    b.y = W[(size_t)(ka + 1) * DFEAT + nBase + col];
    acc = __builtin_amdgcn_wmma_f32_16x16x4_f32(
        /*neg_a=*/false, a, /*neg_b=*/false, b,
        /*c_mod=*/(short)0, acc, /*reuse_a=*/false, /*reuse_b=*/false);
  }

#pragma unroll
  for (int v = 0; v < 8; ++v) {
    const int m = v + khalf * 8;                     // C/D layout: lanes>=16 -> M+8
    H[(size_t)(mBase + m) * DFEAT + nBase + col] = acc[v];
  }
}

// ---------------- aggregation ----------------

// agg[n,:] = dinv[n]^2 * h[n,:]   (self-loop term; also initializes agg)
// float4-vectorized: one thread per 4 features.
__global__ void k_self_init(const float4* __restrict__ h4, const float* __restrict__ dinv,
                            float4* __restrict__ agg4, int total4) {
  int i = blockIdx.x * blockDim.x + threadIdx.x;
  if (i < total4) {
    const int n = i >> 4;                 // 16 float4 per node row
    const float dv = dinv[n];
    const float s = dv * dv;
    float4 v = h4[i];
    v.x *= s; v.y *= s; v.z *= s; v.w *= s;
    agg4[i] = v;
  }
}

// agg[col[e],:] += dinv[row]*dinv[col] * h[row[e],:]
// 16 threads per edge; each thread gathers a float4 (global_load_b128) and
// issues 4 global_atomic_add_f32 (atomic count is irreducible in f32).
__global__ void k_edge_agg(const int* __restrict__ ei, const float* __restrict__ dinv,
                           const float* __restrict__ h, float* __restrict__ agg, int nE) {
  const long long t = (long long)blockIdx.x * blockDim.x + threadIdx.x;
  const int e = (int)(t >> 4);
  if (e >= nE) return;
  const int d4 = (int)(t & 15);
  const int r = ei[e];          // source
  const int c = ei[nE + e];     // target
  const float w = dinv[r] * dinv[c];
  const float4 hv = ((const float4*)(h + (size_t)r * DFEAT))[d4];
  float* dst = agg + (size_t)c * DFEAT + d4 * 4;
  atomAddF(dst + 0, w * hv.x);
  atomAddF(dst + 1, w * hv.y);
  atomAddF(dst + 2, w * hv.z);
  atomAddF(dst + 3, w * hv.w);
}

// z = relu(agg + b), float4-vectorized
__global__ void k_bias_relu(const float4* __restrict__ agg4, const float4* __restrict__ b4,
                            float4* __restrict__ z4, int total4) {
  int i = blockIdx.x * blockDim.x + threadIdx.x;
  if (i < total4) {
    const float4 bb = b4[i & 15];         // 16 float4 per 64-feature row
    float4 v = agg4[i];
    v.x = fmaxf(v.x + bb.x, 0.0f);
    v.y = fmaxf(v.y + bb.y, 0.0f);
    v.z = fmaxf(v.z + bb.z, 0.0f);
    v.w = fmaxf(v.w + bb.w, 0.0f);
    z4[i] = v;
  }
}

// ---------------- launcher ----------------

extern "C" void kernel_launch(void* const* d_in, const int* in_sizes, int n_in,
                              void* d_out, int out_size, void* d_ws, size_t ws_size,
                              hipStream_t stream) {
  const float* x  = (const float*)d_in[0];
  const float* W1 = (const float*)d_in[1];
  const float* b1 = (const float*)d_in[2];
  const float* W2 = (const float*)d_in[3];
  const float* b2 = (const float*)d_in[4];
  const int*   ei = (const int*)d_in[5];   // [2,E] flat: [0:E)=row, [E:2E)=col

  float* out = (float*)d_out;

  // workspace carve-up: dinv (padded) + two N*D fp32 buffers (~25.8 MB total)
  float* dinv = (float*)d_ws;
  float* bufA = dinv + 50176;                        // N padded to 256
  float* bufB = bufA + (size_t)NNODES * DFEAT;

  const int total  = NNODES * DFEAT;
  const int total4 = total / 4;
  const dim3 thr256(256);
  const dim3 gemmBlk(32, 4);                         // 4 waves = 4 N-tiles
  const dim3 gemmGrd(NNODES / 16);                   // 3125 M-tiles
  const unsigned edgeGrid = (unsigned)(((long long)NEDGES * 16 + 255) / 256);
  const unsigned nodeGrid4 = (unsigned)((total4 + 255) / 256);

  // normalization: deg = 1 + indeg; dinv = rsqrt(deg)
  k_set1<<<(NNODES + 255) / 256, thr256, 0, stream>>>(dinv, NNODES);
  k_degcount<<<(NEDGES + 255) / 256, thr256, 0, stream>>>(ei, dinv, NEDGES);
  k_rsqrt_inplace<<<(NNODES + 255) / 256, thr256, 0, stream>>>(dinv, NNODES);

  // layer 1: h1 = x@W1 (bufA); agg1 (bufB); z1 = relu(agg1+b1) (bufA)
  k_gemm_wmma<<<gemmGrd, gemmBlk, 0, stream>>>(x, W1, bufA);
  k_self_init<<<nodeGrid4, thr256, 0, stream>>>((const float4*)bufA, dinv,
                                                (float4*)bufB, total4);
  k_edge_agg<<<edgeGrid, thr256, 0, stream>>>(ei, dinv, bufA, bufB, NEDGES);
  k_bias_relu<<<nodeGrid4, thr256, 0, stream>>>((const float4*)bufB, (const float4*)b1,
                                                (float4*)bufA, total4);

  // layer 2: h2 = z1@W2 (bufB); agg2 directly in d_out; relu in place
  k_gemm_wmma<<<gemmGrd, gemmBlk, 0, stream>>>(bufA, W2, bufB);
  k_self_init<<<nodeGrid4, thr256, 0, stream>>>((const float4*)bufB, dinv,
                                                (float4*)out, total4);
  k_edge_agg<<<edgeGrid, thr256, 0, stream>>>(ei, dinv, bufB, out, NEDGES);
  k_bias_relu<<<nodeGrid4, thr256, 0, stream>>>((const float4*)out, (const float4*)b2,
                                                (float4*)out, total4);
}